// weightup_43980465111581
// MI455X (gfx1250) — compile-verified
//
#include <hip/hip_runtime.h>
#include <math.h>

// ---------------- problem constants ----------------
#define M_DIM   4096
#define NSTEPS  199
#define LR      0.1f
#define B1C     0.9f
#define B2C     0.999f
#define EPSC    1e-08f

// ---------------- tiling constants ----------------
#define KSPLIT        8                 // K split for parallelism (deterministic partials)
#define KCHUNK        (M_DIM / KSPLIT)  // 512
#define WAVES_PER_BLK 8
#define ROWS_PER_BLK  (WAVES_PER_BLK * 16)  // 128
#define ROW_BLKS      (M_DIM / ROWS_PER_BLK) // 32

typedef float v2f __attribute__((ext_vector_type(2)));
typedef float v8f __attribute__((ext_vector_type(8)));

#if defined(__has_builtin)
#if __has_builtin(__builtin_amdgcn_wmma_f32_16x16x4_f32)
#define HAVE_WMMA_F32X4 1
#endif
#endif

// ---------------------------------------------------------------
// Qs = 0.5 * (va + va^T), LDS-tiled so BOTH operands read coalesced.
// grid (128,128), block (32,8)
// ---------------------------------------------------------------
__global__ __launch_bounds__(256) void symmetrize_kernel(const float* __restrict__ va,
                                                         float* __restrict__ Qs) {
    __shared__ float tile[32][33];     // +1 pad: conflict-free transpose
    const int bx = blockIdx.x * 32;    // output column block
    const int by = blockIdx.y * 32;    // output row block
    const int tx = threadIdx.x;        // 0..31
    // stage va[bx + i][by + tx]  (the transposed operand), coalesced
    for (int i = threadIdx.y; i < 32; i += 8)
        tile[i][tx] = va[(size_t)(bx + i) * M_DIM + by + tx];
    __syncthreads();
    // Qs[by+i][bx+tx] = 0.5*(va[by+i][bx+tx] + va[bx+tx][by+i])
    for (int i = threadIdx.y; i < 32; i += 8)
        Qs[(size_t)(by + i) * M_DIM + bx + tx] =
            0.5f * (va[(size_t)(by + i) * M_DIM + bx + tx] + tile[tx][i]);
}

// ---------------------------------------------------------------
// init: w=1, mt=vt=0, u = w - mean
// ---------------------------------------------------------------
__global__ __launch_bounds__(256) void init_kernel(const float* __restrict__ mean,
                                                   float* __restrict__ w,
                                                   float* __restrict__ mt,
                                                   float* __restrict__ vt,
                                                   float* __restrict__ u) {
    const int i = blockIdx.x * blockDim.x + threadIdx.x;
    if (i < M_DIM) {
        w[i]  = 1.0f;
        mt[i] = 0.0f;
        vt[i] = 0.0f;
        u[i]  = 1.0f - mean[i];
    }
}

// ---------------------------------------------------------------
// GEMV partial: g_part[ks][row] = sum_{k in ks-chunk} Qs[row][k] * u[k]
// One wave owns 16 rows. WMMA f32 16x16x4: A = Q 16x4 tile (b64/lane),
// B carries the u chunk in column n==0 only (built branch-free with
// broadcast ds_load_b64 + v_cndmask), C accumulates across K.
// grid (ROW_BLKS, KSPLIT), block 256 (8 waves).
// ---------------------------------------------------------------
__global__ __launch_bounds__(256) void gemv_partial_kernel(const float* __restrict__ Qs,
                                                           const float* __restrict__ u,
                                                           float* __restrict__ g_part) {
    const int lane = threadIdx.x & 31;
    const int wave = threadIdx.x >> 5;
    const int ks   = blockIdx.y;
    const int k0   = ks * KCHUNK;

    // stage this block's u chunk (2 KB) into LDS once
    __shared__ float su[KCHUNK];
    for (int i = threadIdx.x; i < KCHUNK; i += 256) su[i] = u[k0 + i];
    __syncthreads();

    const int row0 = blockIdx.x * ROWS_PER_BLK + wave * 16;
    const int m    = lane & 15;          // row within wave tile
    const int kh   = (lane >> 4) * 2;    // K half: lanes 0-15 -> K{0,1}, 16-31 -> K{2,3}
    const bool sel = (m == 0);           // loop-invariant column-0 mask
    const float* aptr = Qs + (size_t)(row0 + m) * M_DIM + k0 + kh;
    const float* bptr = su + kh;         // lanes 0-15: u[kk..kk+1]; 16-31: u[kk+2..kk+3]

#if HAVE_WMMA_F32X4
    v8f acc = {};  // C/D: 16x16 f32, we use column 0
    #pragma unroll 8
    for (int kk = 0; kk < KCHUNK; kk += 4) {
        // A layout (16x4 f32): lane l: VGPR0 = A[m, kh], VGPR1 = A[m, kh+1]
        v2f a  = *(const v2f*)(aptr + kk);
        // broadcast ds_load_b64 for ALL lanes (no exec predication), then
        // select into column 0 only: compiles to 2x v_cndmask, no SALU churn
        v2f bv = *(const v2f*)(bptr + kk);
        v2f b;
        b.x = sel ? bv.x : 0.0f;
        b.y = sel ? bv.y : 0.0f;
        acc = __builtin_amdgcn_wmma_f32_16x16x4_f32(false, a, false, b,
                                                    (short)0, acc, false, false);
    }
    // D column 0: lane 0 has rows 0..7 in VGPR 0..7, lane 16 has rows 8..15
    if (sel) {
        float* gp = g_part + (size_t)ks * M_DIM + row0 + (lane >> 4) * 8;
        #pragma unroll
        for (int v = 0; v < 8; ++v) gp[v] = acc[v];
    }
#else
    // VALU fallback: lane pair (l, l+16) splits row m's K range; combine via shfl
    float s = 0.0f;
    #pragma unroll 8
    for (int kk = 0; kk < KCHUNK; kk += 4) {
        v2f a  = *(const v2f*)(aptr + kk);
        v2f bv = *(const v2f*)(bptr + kk);
        s += a.x * bv.x + a.y * bv.y;
    }
    s += __shfl_xor(s, 16, 32);
    if (lane < 16) g_part[(size_t)ks * M_DIM + row0 + m] = s;
#endif
}

// ---------------------------------------------------------------
// Adam update: sum K-split partials (fixed order -> deterministic),
// then Adam step; writes w and u = w - mean for next GEMV.
// ---------------------------------------------------------------
__global__ __launch_bounds__(256) void adam_update_kernel(const float* __restrict__ g_part,
                                                          const float* __restrict__ mean,
                                                          float* __restrict__ w,
                                                          float* __restrict__ mt,
                                                          float* __restrict__ vt,
                                                          float* __restrict__ u,
                                                          float lr_over_bc1,
                                                          float rsqrt_bc2) {
    const int i = blockIdx.x * blockDim.x + threadIdx.x;
    if (i >= M_DIM) return;
    float g = 0.0f;
    #pragma unroll
    for (int ks = 0; ks < KSPLIT; ++ks) g += g_part[(size_t)ks * M_DIM + i];
    float m_ = B1C * mt[i] + (1.0f - B1C) * g;
    float v_ = B2C * vt[i] + (1.0f - B2C) * g * g;
    mt[i] = m_;
    vt[i] = v_;
    float denom = sqrtf(v_) * rsqrt_bc2 + EPSC;
    float wn = w[i] - lr_over_bc1 * m_ / denom;
    w[i] = wn;
    u[i] = wn - mean[i];
}

__global__ __launch_bounds__(256) void copy_out_kernel(const float* __restrict__ w,
                                                       float* __restrict__ out) {
    const int i = blockIdx.x * blockDim.x + threadIdx.x;
    if (i < M_DIM) out[i] = w[i];
}

// ---------------------------------------------------------------
// host launcher
// ---------------------------------------------------------------
extern "C" void kernel_launch(void* const* d_in, const int* in_sizes, int n_in,
                              void* d_out, int out_size, void* d_ws, size_t ws_size,
                              hipStream_t stream) {
    const float* mean = (const float*)d_in[0];
    const float* va   = (const float*)d_in[1];
    // d_in[2] (xt) is unused by the reference.
    float* out = (float*)d_out;

    // workspace layout (floats): Qs (64 MB) | g_part | w | mt | vt | u
    float* ws     = (float*)d_ws;
    float* Qs     = ws;
    float* g_part = Qs + (size_t)M_DIM * M_DIM;
    float* w      = g_part + (size_t)KSPLIT * M_DIM;
    float* mt     = w  + M_DIM;
    float* vt     = mt + M_DIM;
    float* u      = vt + M_DIM;

    // 1) Qs = 0.5*(va + va^T), once; resident in 192 MB L2 for all 199 GEMVs
    symmetrize_kernel<<<dim3(M_DIM / 32, M_DIM / 32), dim3(32, 8), 0, stream>>>(va, Qs);

    // 2) optimizer state init
    init_kernel<<<M_DIM / 256, 256, 0, stream>>>(mean, w, mt, vt, u);

    // 3) 199 sequential Adam steps: GEMV partials + fused reduce/update
    for (int t = 1; t <= NSTEPS; ++t) {
        gemv_partial_kernel<<<dim3(ROW_BLKS, KSPLIT), 256, 0, stream>>>(Qs, u, g_part);
        const float bc1 = 1.0f - powf(B1C, (float)t);
        const float bc2 = 1.0f - powf(B2C, (float)t);
        adam_update_kernel<<<M_DIM / 256, 256, 0, stream>>>(
            g_part, mean, w, mt, vt, u, LR / bc1, 1.0f / sqrtf(bc2));
    }

    // 4) final w -> output
    copy_out_kernel<<<M_DIM / 256, 256, 0, stream>>>(w, out);
}